// InvarianceModelVaryingSizes_50740743635791
// MI455X (gfx1250) — compile-verified
//
#include <hip/hip_runtime.h>
#include <hip/hip_bf16.h>

typedef __attribute__((ext_vector_type(2))) float v2f;
typedef __attribute__((ext_vector_type(8))) float v8f;

#define NSEG   8192
#define D_IN   64
#define D_HID  128
#define LDS_PITCH 132   // 128 + 4 pad to break same-bank rows

// ---------------------------------------------------------------------------
// Kernel 0: exclusive prefix sum of segment lengths -> segment start offsets.
// ---------------------------------------------------------------------------
__global__ void seg_scan_kernel(const int* __restrict__ l, int* __restrict__ offs) {
  __shared__ int ps[256];
  const int tid = threadIdx.x;
  int local[32];
  int s = 0;
#pragma unroll
  for (int i = 0; i < 32; ++i) {
    int v = l[tid * 32 + i];
    local[i] = v;
    s += v;
  }
  ps[tid] = s;
  __syncthreads();
  for (int off = 1; off < 256; off <<= 1) {
    int v = 0;
    if (tid >= off) v = ps[tid - off];
    __syncthreads();
    if (tid >= off) ps[tid] += v;
    __syncthreads();
  }
  int run = (tid == 0) ? 0 : ps[tid - 1];
#pragma unroll
  for (int i = 0; i < 32; ++i) {
    offs[tid * 32 + i] = run;
    run += local[i];
  }
}

// 16-token-tile MMA: ks outer, nt inner -> 8 independent WMMA chains.
// a[16] are the per-lane A fragments, bcol points at this lane's B column
// (w1s + 2*half*LDS_PITCH + lm); c[8] accumulates 8 N-tiles.
__device__ __forceinline__ void tile_mma_16x128(const v2f a[16],
                                                const float* __restrict__ bcol,
                                                v8f c[8]) {
#pragma unroll
  for (int ks = 0; ks < 16; ++ks) {
    const float* bp = bcol + (ks * 4) * LDS_PITCH;
#pragma unroll
    for (int nt = 0; nt < 8; ++nt) {
      v2f b;
      b.x = bp[nt * 16];
      b.y = bp[nt * 16 + LDS_PITCH];
      c[nt] = __builtin_amdgcn_wmma_f32_16x16x4_f32(
          false, a[ks], false, b, (short)0, c[nt], false, false);
    }
  }
}

// ---------------------------------------------------------------------------
// Kernel 1: fused  relu(x@W1+b1) + ragged segment-sum  -> xm [NSEG, 128].
// 8 waves/block, wave w owns segment blockIdx.x*8+w; W1 staged in LDS.
// ---------------------------------------------------------------------------
__global__ __launch_bounds__(256) void fused_mlp_segsum_kernel(
    const float* __restrict__ x, const float* __restrict__ W1,
    const float* __restrict__ b1, const int* __restrict__ offs,
    const int* __restrict__ l, float* __restrict__ xm) {
  __shared__ float w1s[D_IN * LDS_PITCH];

  const int tid = threadIdx.x;
  for (int i = tid; i < D_IN * D_HID; i += 256) {
    w1s[(i >> 7) * LDS_PITCH + (i & 127)] = W1[i];
  }
  __syncthreads();

  const int lane = tid & 31;
  const int wave = tid >> 5;
  const int seg  = blockIdx.x * 8 + wave;
  const int start = offs[seg];
  const int len   = l[seg];
  const int half = lane >> 4;   // 0: lanes 0-15 (K+0/1), 1: lanes 16-31 (K+2/3)
  const int lm   = lane & 15;

  const float* bcol = w1s + (2 * half) * LDS_PITCH + lm;

  float bias[8];
#pragma unroll
  for (int nt = 0; nt < 8; ++nt) bias[nt] = b1[nt * 16 + lm];

  float accS[8];
#pragma unroll
  for (int nt = 0; nt < 8; ++nt) accS[nt] = 0.0f;

  const int nfull = len & ~15;
  int t0 = 0;

  // ---- full tiles: no masking anywhere ----
  for (; t0 < nfull; t0 += 16) {
    const float* xrow = x + (size_t)(start + t0 + lm) * D_IN + 2 * half;
    v2f a[16];
#pragma unroll
    for (int ks = 0; ks < 16; ++ks) a[ks] = *(const v2f*)(xrow + ks * 4);

    v8f c[8];
#pragma unroll
    for (int nt = 0; nt < 8; ++nt)
#pragma unroll
      for (int e = 0; e < 8; ++e) c[nt][e] = 0.0f;

    tile_mma_16x128(a, bcol, c);

#pragma unroll
    for (int nt = 0; nt < 8; ++nt) {
      float v[8];
#pragma unroll
      for (int r = 0; r < 8; ++r) v[r] = fmaxf(c[nt][r] + bias[nt], 0.0f);
      float s = ((v[0] + v[1]) + (v[2] + v[3])) + ((v[4] + v[5]) + (v[6] + v[7]));
      accS[nt] += s;
    }
  }

  // ---- tail tile: mask token loads and row contributions ----
  if (t0 < len) {
    const bool tokv = (t0 + lm) < len;
    const float* xrow = x + (size_t)(start + t0 + lm) * D_IN + 2 * half;
    v2f a[16];
#pragma unroll
    for (int ks = 0; ks < 16; ++ks) {
      v2f av;
      av.x = 0.0f;
      av.y = 0.0f;
      if (tokv) av = *(const v2f*)(xrow + ks * 4);
      a[ks] = av;
    }

    v8f c[8];
#pragma unroll
    for (int nt = 0; nt < 8; ++nt)
#pragma unroll
      for (int e = 0; e < 8; ++e) c[nt][e] = 0.0f;

    tile_mma_16x128(a, bcol, c);

#pragma unroll
    for (int nt = 0; nt < 8; ++nt) {
      float v[8];
#pragma unroll
      for (int r = 0; r < 8; ++r) {
        const bool rv = (t0 + r + 8 * half) < len;   // dead rows must not add relu(b1)
        v[r] = rv ? fmaxf(c[nt][r] + bias[nt], 0.0f) : 0.0f;
      }
      float s = ((v[0] + v[1]) + (v[2] + v[3])) + ((v[4] + v[5]) + (v[6] + v[7]));
      accS[nt] += s;
    }
  }

  // fold rows 0-7 (lanes 0-15) with rows 8-15 (lanes 16-31), then store.
#pragma unroll
  for (int nt = 0; nt < 8; ++nt) {
    accS[nt] += __shfl_xor(accS[nt], 16, 32);
  }
  if (lane < 16) {
#pragma unroll
    for (int nt = 0; nt < 8; ++nt) {
      xm[(size_t)seg * D_HID + nt * 16 + lane] = accS[nt];
    }
  }
}

// ---------------------------------------------------------------------------
// Kernel 2: out = xm @ W2 + b2   (8192x128 @ 128x128), f32 WMMA, K=128.
// ks outer / nt inner: 8 independent accumulator chains, low live-range.
// ---------------------------------------------------------------------------
__global__ __launch_bounds__(32) void head_gemm_kernel(
    const float* __restrict__ xm, const float* __restrict__ W2,
    const float* __restrict__ b2, float* __restrict__ out) {
  const int lane = threadIdx.x & 31;
  const int half = lane >> 4;
  const int lm   = lane & 15;
  const int mt   = blockIdx.x;  // 0..511

  const float* arow = xm + (size_t)(mt * 16 + lm) * D_HID + 2 * half;
  const float* bcol = W2 + (2 * half) * D_HID + lm;

  v8f c[8];
#pragma unroll
  for (int nt = 0; nt < 8; ++nt)
#pragma unroll
    for (int e = 0; e < 8; ++e) c[nt][e] = 0.0f;

#pragma unroll
  for (int ks = 0; ks < 32; ++ks) {
    v2f a = *(const v2f*)(arow + ks * 4);
    const float* bp = bcol + (ks * 4) * D_HID;
#pragma unroll
    for (int nt = 0; nt < 8; ++nt) {
      v2f b;
      b.x = bp[nt * 16];
      b.y = bp[nt * 16 + D_HID];
      c[nt] = __builtin_amdgcn_wmma_f32_16x16x4_f32(
          false, a, false, b, (short)0, c[nt], false, false);
    }
  }

#pragma unroll
  for (int nt = 0; nt < 8; ++nt) {
    const int n = nt * 16 + lm;
    const float bias = b2[n];
#pragma unroll
    for (int r = 0; r < 8; ++r) {
      out[(size_t)(mt * 16 + r + 8 * half) * D_HID + n] = c[nt][r] + bias;
    }
  }
}

// ---------------------------------------------------------------------------
extern "C" void kernel_launch(void* const* d_in, const int* in_sizes, int n_in,
                              void* d_out, int out_size, void* d_ws, size_t ws_size,
                              hipStream_t stream) {
  (void)in_sizes; (void)n_in; (void)out_size; (void)ws_size;
  const float* x  = (const float*)d_in[0];
  const int*   l  = (const int*)d_in[1];
  const float* W1 = (const float*)d_in[2];
  const float* b1 = (const float*)d_in[3];
  const float* W2 = (const float*)d_in[4];
  const float* b2 = (const float*)d_in[5];
  float* out = (float*)d_out;

  int*   offs = (int*)d_ws;
  float* xm   = (float*)((char*)d_ws + ((NSEG * sizeof(int) + 255) & ~size_t(255)));

  seg_scan_kernel<<<1, 256, 0, stream>>>(l, offs);
  fused_mlp_segsum_kernel<<<NSEG / 8, 256, 0, stream>>>(x, W1, b1, offs, l, xm);
  head_gemm_kernel<<<NSEG / 16, 32, 0, stream>>>(xm, W2, b2, out);
}